// HMM_27960237097352
// MI455X (gfx1250) — compile-verified
//
#include <hip/hip_runtime.h>
#include <math.h>

// HMM forward filter, M=1024 states, T=8192 steps, D=8 obs dim.
// Phase 1: row softmax -> Gamma (ws)
// Phase 2: emission densities g (T x M) via V_WMMA_F32_16X16X4_F32
// Phase 3: persistent 16-block cooperative scan, Gamma column-partitioned
//          into 264KB of LDS per WGP (CDNA5 320KB LDS), one L2-atomic
//          grid sync per step.

#define Msz 1024
#define Tsz 8192
#define Dsz 8
#define SCAN_BLOCKS 16
#define SCAN_THREADS 512
#define COLS_PER_BLOCK 64
#define GSTRIDE 1032   // 64-col LDS row stride (floats); 1032%64==8 -> conflict-free

typedef __attribute__((ext_vector_type(2))) float v2f;
typedef __attribute__((ext_vector_type(8))) float v8f;

// -0.5 * D * log(2*pi), D=8
#define NEG_HALF_D_LOG2PI (-7.3515082656373813f)

// ---------------------------------------------------------------------------
// Prep: normalize pi, musq, ysq, zero sync counter + fparts. Single block.
// ---------------------------------------------------------------------------
__global__ __launch_bounds__(256) void hmm_prep(const float* __restrict__ init,
                                                const float* __restrict__ mu,
                                                const float* __restrict__ y,
                                                float* __restrict__ pi,
                                                float* __restrict__ musq,
                                                float* __restrict__ ysq,
                                                unsigned* __restrict__ counter,
                                                float* __restrict__ fparts) {
  __shared__ float sred[256];
  const int tid = threadIdx.x;
  float s = 0.f;
  for (int i = tid; i < Msz; i += 256) s += init[i];
  sred[tid] = s;
  __syncthreads();
  for (int o = 128; o > 0; o >>= 1) {
    if (tid < o) sred[tid] += sred[tid + o];
    __syncthreads();
  }
  const float inv = 1.0f / sred[0];
  for (int i = tid; i < Msz; i += 256) pi[i] = init[i] * inv;
  for (int m = tid; m < Msz; m += 256) {
    float q = 0.f;
    #pragma unroll
    for (int d = 0; d < Dsz; ++d) { float v = mu[m * Dsz + d]; q = fmaf(v, v, q); }
    musq[m] = q;
  }
  for (int t = tid; t < Tsz; t += 256) {
    float q = 0.f;
    #pragma unroll
    for (int d = 0; d < Dsz; ++d) { float v = y[t * Dsz + d]; q = fmaf(v, v, q); }
    ysq[t] = q;
  }
  if (tid == 0) *counter = 0u;
  if (tid < 32) fparts[tid] = 0.f;
}

// ---------------------------------------------------------------------------
// Row softmax over [logits_row (1023), implicit 0] -> Gamma row (1024).
// One block per row.
// ---------------------------------------------------------------------------
__global__ __launch_bounds__(256) void hmm_softmax(const float* __restrict__ logits,
                                                   float* __restrict__ Gamma) {
  __shared__ float sred[256];
  const int r = blockIdx.x;
  const int tid = threadIdx.x;
  const float* row = logits + (size_t)r * (Msz - 1);

  float mx = 0.0f;  // implicit zero logit participates in the max
  for (int j = tid; j < Msz - 1; j += 256) mx = fmaxf(mx, row[j]);
  sred[tid] = mx;
  __syncthreads();
  for (int o = 128; o > 0; o >>= 1) {
    if (tid < o) sred[tid] = fmaxf(sred[tid], sred[tid + o]);
    __syncthreads();
  }
  mx = sred[0];
  __syncthreads();

  float s = (tid == 0) ? expf(-mx) : 0.0f;  // implicit zero logit term
  for (int j = tid; j < Msz - 1; j += 256) s += expf(row[j] - mx);
  sred[tid] = s;
  __syncthreads();
  for (int o = 128; o > 0; o >>= 1) {
    if (tid < o) sred[tid] += sred[tid + o];
    __syncthreads();
  }
  const float invs = 1.0f / sred[0];

  float* grow = Gamma + (size_t)r * Msz;
  for (int j = tid; j < Msz - 1; j += 256) grow[j] = expf(row[j] - mx) * invs;
  if (tid == 0) grow[Msz - 1] = expf(-mx) * invs;
}

// ---------------------------------------------------------------------------
// Emission densities via WMMA f32 16x16x4.
//   WMMA-M = state tile (mu rows), WMMA-N = time tile (y rows), K = D chunks.
//   D[m][t] = sum_k mu[m][k]*y[t][k]; g = exp(K0 - 0.5*ysq - 0.5*musq + dot)
// 8 waves/block, one 16x16 tile per wave.
// ---------------------------------------------------------------------------
__global__ __launch_bounds__(256) void hmm_emission(const float* __restrict__ mu,
                                                    const float* __restrict__ y,
                                                    const float* __restrict__ musq,
                                                    const float* __restrict__ ysq,
                                                    float* __restrict__ g) {
  const int lane = threadIdx.x & 31;
  const int wave = threadIdx.x >> 5;            // 0..7
  const int m0 = blockIdx.x * 16;               // state tile
  const int t0 = (blockIdx.y * 8 + wave) * 16;  // time tile
  const int l = lane & 15;
  const int half = lane >> 4;

  v8f acc = {};
  // K chunk 0: K = 0..3 ; K chunk 1: K = 4..7
  #pragma unroll
  for (int c = 0; c < 2; ++c) {
    const int ka = c * 4 + half * 2;
    v2f a, b;
    // A (16x4 f32): lane l holds row M=l; lanes 0-15 -> K0/K1, 16-31 -> K2/K3
    a.x = mu[(size_t)(m0 + l) * Dsz + ka];
    a.y = mu[(size_t)(m0 + l) * Dsz + ka + 1];
    // B (4x16 f32): lane l holds col N=l; VGPR0/1 -> K rows (half selects K+2)
    b.x = y[(size_t)(t0 + l) * Dsz + ka];
    b.y = y[(size_t)(t0 + l) * Dsz + ka + 1];
    acc = __builtin_amdgcn_wmma_f32_16x16x4_f32(false, a, false, b,
                                                (short)0, acc, false, false);
  }

  const int t = t0 + l;                 // D-matrix: N = lane&15
  const int mbase = m0 + half * 8;      // D-matrix: M = r + 8*half
  const float base = NEG_HALF_D_LOG2PI - 0.5f * ysq[t];

  float4 o0, o1;
  o0.x = expf(base + acc[0] - 0.5f * musq[mbase + 0]);
  o0.y = expf(base + acc[1] - 0.5f * musq[mbase + 1]);
  o0.z = expf(base + acc[2] - 0.5f * musq[mbase + 2]);
  o0.w = expf(base + acc[3] - 0.5f * musq[mbase + 3]);
  o1.x = expf(base + acc[4] - 0.5f * musq[mbase + 4]);
  o1.y = expf(base + acc[5] - 0.5f * musq[mbase + 5]);
  o1.z = expf(base + acc[6] - 0.5f * musq[mbase + 6]);
  o1.w = expf(base + acc[7] - 0.5f * musq[mbase + 7]);

  *(float4*)&g[(size_t)t * Msz + mbase]     = o0;
  *(float4*)&g[(size_t)t * Msz + mbase + 4] = o1;
}

// ---------------------------------------------------------------------------
// Persistent cooperative scan. 16 blocks x 512 threads, 1 block per WGP.
// Block b owns output columns [64b, 64b+64). Gamma columns live in LDS.
// One grid sync per step via monotonic L2 atomic counter.
// ---------------------------------------------------------------------------
__global__ __launch_bounds__(SCAN_THREADS, 1) void hmm_scan(
    const float* __restrict__ Gamma, const float* __restrict__ g,
    const float* __restrict__ pi, float* __restrict__ Ut, float* __restrict__ ft,
    float* __restrict__ Utt, float* fparts, unsigned* counter) {
  __shared__ float Glds[COLS_PER_BLOCK * GSTRIDE];  // 264,192 B
  __shared__ float u_lds[Msz];                      // 4 KB
  __shared__ float u_stage[COLS_PER_BLOCK];
  __shared__ float red2[2];
  __shared__ float f_sh;

  const int tid = threadIdx.x;
  const int b = blockIdx.x;
  const int col0 = b * COLS_PER_BLOCK;

  // Load this block's Gamma columns into LDS (coalesced 64-float segments).
  for (int idx = tid; idx < COLS_PER_BLOCK * Msz; idx += SCAN_THREADS) {
    const int jl = idx & (COLS_PER_BLOCK - 1);
    const int i  = idx >> 6;
    Glds[jl * GSTRIDE + i] = Gamma[(size_t)i * Msz + col0 + jl];
  }
  for (int i = tid; i < Msz; i += SCAN_THREADS) u_lds[i] = pi[i];
  __syncthreads();

  const int j_local = tid >> 3;   // 0..63: column within block
  const int sub     = tid & 7;    // reduction slice
  const float4* gcol  = reinterpret_cast<const float4*>(&Glds[j_local * GSTRIDE + sub * 128]);
  const float4* ubase = reinterpret_cast<const float4*>(&u_lds[sub * 128]);

  for (int t = 0; t < Tsz; ++t) {
    // u_t[j] = sum_i u_prev[i] * Gamma[i][j]  (each thread: 128-elem slice)
    float p = 0.f;
    #pragma unroll 8
    for (int k = 0; k < 32; ++k) {
      const float4 a = gcol[k];
      const float4 u = ubase[k];
      p = fmaf(a.x, u.x, fmaf(a.y, u.y, fmaf(a.z, u.z, fmaf(a.w, u.w, p))));
    }
    // reduce 8 slices (contiguous lane groups within the wave32)
    p += __shfl_xor(p, 1);
    p += __shfl_xor(p, 2);
    p += __shfl_xor(p, 4);
    if (sub == 0) {
      u_stage[j_local] = p;
      Utt[(size_t)t * Msz + col0 + j_local] = p;  // predicted posterior output
    }
    __syncthreads();

    // block-partial normalizer: sum_j u_t[j]*g[t][j] over our 64 columns
    if (tid < 64) {
      float v = u_stage[tid] * g[(size_t)t * Msz + col0 + tid];
      v += __shfl_xor(v, 1);
      v += __shfl_xor(v, 2);
      v += __shfl_xor(v, 4);
      v += __shfl_xor(v, 8);
      v += __shfl_xor(v, 16);
      if ((tid & 31) == 0) red2[tid >> 5] = v;
    }
    __syncthreads();
    if (tid == 0) fparts[(t & 1) * SCAN_BLOCKS + b] = red2[0] + red2[1];

    // ---- grid sync (release): all global stores visible before arrival ----
    __threadfence();
    __syncthreads();
    if (tid == 0) {
      __hip_atomic_fetch_add(counter, 1u, __ATOMIC_RELEASE, __HIP_MEMORY_SCOPE_AGENT);
      const unsigned target = (unsigned)SCAN_BLOCKS * (unsigned)(t + 1);
      while (__hip_atomic_load(counter, __ATOMIC_ACQUIRE, __HIP_MEMORY_SCOPE_AGENT) < target) {
        __builtin_amdgcn_s_sleep(2);
      }
    }
    __syncthreads();
    __threadfence();  // acquire side: invalidate before cross-block reads

    if (tid == 0) {
      float f = 0.f;
      const float* fp = &fparts[(t & 1) * SCAN_BLOCKS];
      #pragma unroll
      for (int q = 0; q < SCAN_BLOCKS; ++q) f += fp[q];
      f_sh = f;
      if (b == 0) ft[t] = f;
    }
    __syncthreads();

    // filtered posterior for next step: u_tt = u_t * g_t / f_t (full vector,
    // computed redundantly per block; we own stores for our 64 columns)
    const float invf = 1.0f / f_sh;
    for (int i = tid; i < Msz; i += SCAN_THREADS) {
      const float up = Utt[(size_t)t * Msz + i];
      const float un = up * g[(size_t)t * Msz + i] * invf;
      u_lds[i] = un;
      if ((i >> 6) == b) Ut[(size_t)t * Msz + i] = un;
    }
    __syncthreads();
  }
}

// ---------------------------------------------------------------------------
extern "C" void kernel_launch(void* const* d_in, const int* in_sizes, int n_in,
                              void* d_out, int out_size, void* d_ws, size_t ws_size,
                              hipStream_t stream) {
  (void)in_sizes; (void)n_in; (void)out_size; (void)ws_size;
  const float* logits = (const float*)d_in[0];  // (1024, 1023)
  const float* init   = (const float*)d_in[1];  // (1024,)
  const float* mu     = (const float*)d_in[2];  // (1024, 8)
  const float* y      = (const float*)d_in[3];  // (8192, 8)

  float* Ut  = (float*)d_out;                   // (T, M) filtered
  float* ft  = Ut + (size_t)Tsz * Msz;          // (T,)
  float* Utt = ft + Tsz;                        // (T, M) predicted

  char* ws = (char*)d_ws;
  float* Gamma = (float*)ws;  ws += sizeof(float) * (size_t)Msz * Msz;
  float* g     = (float*)ws;  ws += sizeof(float) * (size_t)Tsz * Msz;
  float* pi    = (float*)ws;  ws += sizeof(float) * Msz;
  float* musq  = (float*)ws;  ws += sizeof(float) * Msz;
  float* ysq   = (float*)ws;  ws += sizeof(float) * Tsz;
  float* fparts = (float*)ws; ws += sizeof(float) * 32;
  unsigned* counter = (unsigned*)ws;

  hmm_prep<<<dim3(1), dim3(256), 0, stream>>>(init, mu, y, pi, musq, ysq, counter, fparts);
  hmm_softmax<<<dim3(Msz), dim3(256), 0, stream>>>(logits, Gamma);
  hmm_emission<<<dim3(Msz / 16, Tsz / 128), dim3(256), 0, stream>>>(mu, y, musq, ysq, g);
  hmm_scan<<<dim3(SCAN_BLOCKS), dim3(SCAN_THREADS), 0, stream>>>(
      Gamma, g, pi, Ut, ft, Utt, fparts, counter);
}